// VN_MLP_85495618995037
// MI455X (gfx1250) — compile-verified
//
#include <hip/hip_runtime.h>
#include <math.h>

typedef __attribute__((ext_vector_type(2))) float v2f;
typedef __attribute__((ext_vector_type(8))) float v8f;

#define BB 4
#define NN 4096
#define KNB 16
#define SP (BB*NN*KNB)      // 262144 spatial sites (b,n,k)
#define EPSV 1e-6f
#define BNEPS 1e-5f

// fp32 WMMA, D = A(16x4) * B(4x16) + C, wave32.
__device__ __forceinline__ v8f wmma4(v2f a, v2f b, v8f c) {
  return __builtin_amdgcn_wmma_f32_16x16x4_f32(false, a, false, b, (short)0, c,
                                               false, false);
}

// hardware approx (v_sqrt_f32 / v_rcp_f32, ~1 ulp) — skip IEEE fixup sequences
__device__ __forceinline__ float fsqrt(float x) { return __builtin_amdgcn_sqrtf(x); }
__device__ __forceinline__ float frcp(float x)  { return __builtin_amdgcn_rcpf(x); }

// lane-XOR butterfly add via DS_SWIZZLE_B32 (group-of-32: and=0x1f, or=0, xor=X)
#define SWIZ_ADD(v, X) \
  (v) += __int_as_float(__builtin_amdgcn_ds_swizzle(__float_as_int(v), ((X) << 10) | 0x1f))

// ---------------------------------------------------------------- KNN top-16
__global__ __launch_bounds__(256)
void knn_kernel(const float* __restrict__ x, int* __restrict__ idx) {
  __shared__ float sx[256], sy[256], sz[256], sw[256];
  const int b = blockIdx.y;
  const int n = blockIdx.x * 256 + threadIdx.x;
  const float* pb = x + (size_t)b * NN * 3;
  const float qx = pb[n*3+0], qy = pb[n*3+1], qz = pb[n*3+2];
  const float qq = qx*qx + qy*qy + qz*qz;
  float bd[KNB]; int bi[KNB];
#pragma unroll
  for (int j = 0; j < KNB; ++j) { bd[j] = -1e30f; bi[j] = 0; }
  for (int t = 0; t < NN; t += 256) {
    __syncthreads();
    {
      const int m = t + threadIdx.x;
      const float px = pb[m*3+0], py = pb[m*3+1], pz = pb[m*3+2];
      sx[threadIdx.x] = px;
      sy[threadIdx.x] = py;
      sz[threadIdx.x] = pz;
      sw[threadIdx.x] = px*px + py*py + pz*pz;
    }
    __syncthreads();
    for (int j = 0; j < 256; ++j) {
      const float nd = 2.f*(qx*sx[j] + qy*sy[j] + qz*sz[j]) - qq - sw[j];
      if (nd > bd[KNB-1]) {
        bd[KNB-1] = nd; bi[KNB-1] = t + j;
#pragma unroll
        for (int q = KNB-1; q > 0; --q) {
          if (bd[q] > bd[q-1]) {
            float td = bd[q]; bd[q] = bd[q-1]; bd[q-1] = td;
            int   ti = bi[q]; bi[q] = bi[q-1]; bi[q-1] = ti;
          } else break;
        }
      }
    }
  }
  int* op = idx + (size_t)(b*NN + n) * KNB;
#pragma unroll
  for (int j = 0; j < KNB; ++j) op[j] = bi[j];
}

// --------------------------------------------- edge features [c=3][d=3][SP]
__global__ __launch_bounds__(256)
void feat_kernel(const float* __restrict__ x, const int* __restrict__ idx,
                 float* __restrict__ P) {
  const int s = blockIdx.x * 256 + threadIdx.x;   // s = (b*NN + n)*KNB + k
  const int b = s >> 16;
  const int n = (s >> 4) & (NN - 1);
  const float* pb = x + (size_t)b * NN * 3;
  const int m = idx[s];
  const float ax = pb[m*3+0], ay = pb[m*3+1], az = pb[m*3+2];  // neighbor
  const float xx = pb[n*3+0], xy = pb[n*3+1], xz = pb[n*3+2];  // center
  const float cx = ay*xz - az*xy;                               // nb x xb
  const float cy = az*xx - ax*xz;
  const float cz = ax*xy - ay*xx;
  P[(0*3+0)*SP + s] = ax - xx;
  P[(0*3+1)*SP + s] = ay - xy;
  P[(0*3+2)*SP + s] = az - xz;
  P[(1*3+0)*SP + s] = xx;
  P[(1*3+1)*SP + s] = xy;
  P[(1*3+2)*SP + s] = xz;
  P[(2*3+0)*SP + s] = cx;
  P[(2*3+1)*SP + s] = cy;
  P[(2*3+2)*SP + s] = cz;
}

__global__ void zero_stats_kernel(float* stats) { stats[threadIdx.x] = 0.f; }

// ------------------- K1: WMMA linear (3 accumulators over d) + norm stats
// X: [CIN][3][SP], W: [Cout][CIN], H: [Cout][3][SP], stats: sum[64], sumsq[64]
template <int CIN>
__global__ __launch_bounds__(128)
void vn_linear_stats_kernel(const float* __restrict__ X, const float* __restrict__ W,
                            float* __restrict__ H, float* __restrict__ stats) {
  __shared__ float lsum[16], lsq[16];
  const int tid = threadIdx.x;
  if (tid < 16) { lsum[tid] = 0.f; lsq[tid] = 0.f; }
  __syncthreads();
  const int lane = tid & 31, wv = tid >> 5;
  const int hf = lane >> 4, lx = lane & 15;
  const int col = blockIdx.x * 64 + wv * 16 + lx;   // spatial site
  const int mbase = blockIdx.y * 16;
  const int arow = mbase + lx;
  v8f acc0 = {}, acc1 = {}, acc2 = {};
  for (int kk = 0; kk < CIN; kk += 4) {
    const int c0 = kk + 2*hf, c1 = c0 + 1;
    const int cc0 = (c0 < CIN) ? c0 : (CIN - 1);    // clamped addresses
    const int cc1 = (c1 < CIN) ? c1 : (CIN - 1);
    const float m0 = (c0 < CIN) ? 1.f : 0.f;        // VALU masks (fold away
    const float m1 = (c1 < CIN) ? 1.f : 0.f;        // when provable)
    v2f a;
    a.x = W[arow*CIN + cc0] * m0;
    a.y = W[arow*CIN + cc1] * m1;
    const size_t o0 = (size_t)cc0 * 3 * SP + col;
    const size_t o1 = (size_t)cc1 * 3 * SP + col;
    v2f bd0, bd1, bd2;
    bd0.x = X[o0 + 0*SP] * m0;  bd0.y = X[o1 + 0*SP] * m1;
    bd1.x = X[o0 + 1*SP] * m0;  bd1.y = X[o1 + 1*SP] * m1;
    bd2.x = X[o0 + 2*SP] * m0;  bd2.y = X[o1 + 2*SP] * m1;
    acc0 = wmma4(a, bd0, acc0);
    acc1 = wmma4(a, bd1, acc1);
    acc2 = wmma4(a, bd2, acc2);
  }
#pragma unroll
  for (int r = 0; r < 8; ++r) {
    const int ch = mbase + r + 8*hf;
    const float h0 = acc0[r], h1 = acc1[r], h2 = acc2[r];
    const size_t ho = (size_t)ch * 3 * SP + col;
    H[ho] = h0; H[ho + SP] = h1; H[ho + 2*SP] = h2;
    const float nn = fsqrt(h0*h0 + h1*h1 + h2*h2) + EPSV;
    float s1 = nn, s2 = nn*nn;
    SWIZ_ADD(s1, 1);  SWIZ_ADD(s2, 1);   // 16-lane butterfly: lanes with same
    SWIZ_ADD(s1, 2);  SWIZ_ADD(s2, 2);   // (r, half) sum their 16 sites
    SWIZ_ADD(s1, 4);  SWIZ_ADD(s2, 4);
    SWIZ_ADD(s1, 8);  SWIZ_ADD(s2, 8);
    if (lx == 0) { atomicAdd(&lsum[r + 8*hf], s1); atomicAdd(&lsq[r + 8*hf], s2); }
  }
  __syncthreads();
  if (tid < 16) {
    atomicAdd(&stats[mbase + tid], lsum[tid]);
    atomicAdd(&stats[64 + mbase + tid], lsq[tid]);
  }
}

// ------------------------------ K2: per-channel mean / rsqrt(var + eps)
__global__ void finalize_stats_kernel(const float* __restrict__ stats,
                                      float* __restrict__ murs, int Cout) {
  const int c = threadIdx.x;
  if (c >= Cout) return;
  const float rcnt = 1.f / (float)SP;
  const float mu = stats[c] * rcnt;
  const float var = stats[64 + c] * rcnt - mu*mu;
  murs[c] = mu;
  murs[64 + c] = rsqrtf(var + BNEPS);
}

// -------- K3: BN folded into B-operand of WMMA GEMM with D, + VN-LeakyReLU
// H: [CMAT][3][SP] (pre-BN linear output), D: [CMAT][CMAT]
// FINAL=0 -> out [CMAT][3][SP];  FINAL=1 -> out [B,64,N,3,K]
template <int CMAT, int FINAL>
__global__ __launch_bounds__(128)
void vn_bn_relu_kernel(const float* __restrict__ H, const float* __restrict__ D,
                       const float* __restrict__ murs, const float* __restrict__ gam,
                       const float* __restrict__ bet, float* __restrict__ out) {
  const int tid = threadIdx.x;
  const int lane = tid & 31, wv = tid >> 5;
  const int hf = lane >> 4, lx = lane & 15;
  const int col = blockIdx.x * 64 + wv * 16 + lx;
  const int mbase = blockIdx.y * 16;
  const int arow = mbase + lx;
  v8f acc0 = {}, acc1 = {}, acc2 = {};
  for (int kk = 0; kk < CMAT; kk += 4) {
    const int c0 = kk + 2*hf, c1 = c0 + 1;            // provably < CMAT
    v2f a;
    a.x = D[arow*CMAT + c0];
    a.y = D[arow*CMAT + c1];
    const size_t o0 = (size_t)c0 * 3 * SP + col;
    const size_t o1 = (size_t)c1 * 3 * SP + col;
    const float h00 = H[o0], h01 = H[o0 + SP], h02 = H[o0 + 2*SP];
    const float h10 = H[o1], h11 = H[o1 + SP], h12 = H[o1 + 2*SP];
    const float n0 = fsqrt(h00*h00 + h01*h01 + h02*h02) + EPSV;
    const float n1 = fsqrt(h10*h10 + h11*h11 + h12*h12) + EPSV;
    const float f0 = ((n0 - murs[c0]) * murs[64+c0] * gam[c0] + bet[c0]) * frcp(n0);
    const float f1 = ((n1 - murs[c1]) * murs[64+c1] * gam[c1] + bet[c1]) * frcp(n1);
    v2f b0, b1, b2;
    b0.x = h00*f0; b0.y = h10*f1;
    b1.x = h01*f0; b1.y = h11*f1;
    b2.x = h02*f0; b2.y = h12*f1;
    acc0 = wmma4(a, b0, acc0);
    acc1 = wmma4(a, b1, acc1);
    acc2 = wmma4(a, b2, acc2);
  }
#pragma unroll
  for (int r = 0; r < 8; ++r) {
    const int m = mbase + r + 8*hf;
    const size_t mo = (size_t)m * 3 * SP + col;
    const float hm0 = H[mo], hm1 = H[mo + SP], hm2 = H[mo + 2*SP];
    const float nm = fsqrt(hm0*hm0 + hm1*hm1 + hm2*hm2) + EPSV;
    const float fm = ((nm - murs[m]) * murs[64+m] * gam[m] + bet[m]) * frcp(nm);
    const float xb0 = hm0*fm, xb1 = hm1*fm, xb2 = hm2*fm;  // BN output at (m,col)
    const float d0 = acc0[r], d1 = acc1[r], d2 = acc2[r];  // direction D*xbn
    const float dot = xb0*d0 + xb1*d1 + xb2*d2;
    float o0 = xb0, o1 = xb1, o2 = xb2;
    if (dot < 0.f) {
      const float t = dot * frcp(d0*d0 + d1*d1 + d2*d2 + EPSV);
      o0 -= t*d0; o1 -= t*d1; o2 -= t*d2;
    }
    if (FINAL) {
      const int k = col & 15;
      const int n = (col >> 4) & (NN - 1);
      const int b = col >> 16;
      const size_t oo = (((size_t)b*64 + m)*NN + n) * 48 + k;  // [B,64,N,3,K]
      out[oo] = o0; out[oo + 16] = o1; out[oo + 32] = o2;
    } else {
      out[mo] = o0; out[mo + SP] = o1; out[mo + 2*SP] = o2;
    }
  }
}

extern "C" void kernel_launch(void* const* d_in, const int* in_sizes, int n_in,
                              void* d_out, int out_size, void* d_ws, size_t ws_size,
                              hipStream_t stream) {
  (void)in_sizes; (void)n_in; (void)out_size; (void)ws_size;
  const float* x  = (const float*)d_in[0];
  const float* W0 = (const float*)d_in[1];
  const float* g0 = (const float*)d_in[2];
  const float* b0 = (const float*)d_in[3];
  const float* D0 = (const float*)d_in[4];
  const float* W1 = (const float*)d_in[5];
  const float* g1 = (const float*)d_in[6];
  const float* b1 = (const float*)d_in[7];
  const float* D1 = (const float*)d_in[8];
  const float* W2 = (const float*)d_in[9];
  const float* g2 = (const float*)d_in[10];
  const float* b2 = (const float*)d_in[11];
  const float* D2 = (const float*)d_in[12];
  float* out = (float*)d_out;

  // workspace: idx (1 MiB) | P 32ch (96 MiB) | Q 64ch (192 MiB) | stats
  char* ws = (char*)d_ws;
  int*   idx   = (int*)ws;
  float* P     = (float*)(ws + (size_t)1024*1024);
  float* Q     = (float*)(ws + (size_t)1024*1024 + (size_t)32*3*SP*4);
  float* stats = (float*)(ws + (size_t)1024*1024 + (size_t)32*3*SP*4
                             + (size_t)64*3*SP*4);
  float* murs  = stats + 128;

  knn_kernel<<<dim3(NN/256, BB), dim3(256), 0, stream>>>(x, idx);
  feat_kernel<<<dim3(SP/256), dim3(256), 0, stream>>>(x, idx, P);

  const int gx = SP / 64;  // 4096 column tiles (4 waves x 16 cols per block)

  // Block 0: 3 -> 32
  zero_stats_kernel<<<1, 128, 0, stream>>>(stats);
  vn_linear_stats_kernel<3><<<dim3(gx, 2), dim3(128), 0, stream>>>(P, W0, Q, stats);
  finalize_stats_kernel<<<1, 64, 0, stream>>>(stats, murs, 32);
  vn_bn_relu_kernel<32, 0><<<dim3(gx, 2), dim3(128), 0, stream>>>(Q, D0, murs, g0, b0, P);

  // Block 1: 32 -> 32
  zero_stats_kernel<<<1, 128, 0, stream>>>(stats);
  vn_linear_stats_kernel<32><<<dim3(gx, 2), dim3(128), 0, stream>>>(P, W1, Q, stats);
  finalize_stats_kernel<<<1, 64, 0, stream>>>(stats, murs, 32);
  vn_bn_relu_kernel<32, 0><<<dim3(gx, 2), dim3(128), 0, stream>>>(Q, D1, murs, g1, b1, P);

  // Block 2: 32 -> 64, writes transposed final output directly
  zero_stats_kernel<<<1, 128, 0, stream>>>(stats);
  vn_linear_stats_kernel<32><<<dim3(gx, 4), dim3(128), 0, stream>>>(P, W2, Q, stats);
  finalize_stats_kernel<<<1, 64, 0, stream>>>(stats, murs, 64);
  vn_bn_relu_kernel<64, 1><<<dim3(gx, 4), dim3(128), 0, stream>>>(Q, D2, murs, g2, b2, out);
}